// DGCNLayer_8323646620425
// MI455X (gfx1250) — compile-verified
//
#include <hip/hip_runtime.h>

typedef __attribute__((ext_vector_type(2))) float v2f;
typedef __attribute__((ext_vector_type(8))) float v8f;

#define FEAT 128
#define TILE_M 16
#define ALPHA 0.1f

__device__ __forceinline__ v8f wmma_f32(v2f a, v2f b, v8f c) {
  // V_WMMA_F32_16X16X4_F32 : D = A(16x4) * B(4x16) + C, full fp32
  return __builtin_amdgcn_wmma_f32_16x16x4_f32(
      /*neg_a=*/false, a, /*neg_b=*/false, b,
      /*c_mod=*/(short)0, c, /*reuse_a=*/false, /*reuse_b=*/false);
}

// C[M x 128] (op)= A[M x 128] @ W[128 x 128]
// ACC:   C += result (reads prior C)
// FINAL: apply bias + relu on store
template <bool ACC, bool FINAL>
__global__ void __launch_bounds__(256) gemm128(const float* __restrict__ A,
                                               const float* __restrict__ W,
                                               const float* __restrict__ bias,
                                               float* __restrict__ C, int M) {
  __shared__ float sW[FEAT * FEAT];    // W transposed: sW[n*128 + k]  (64 KB)
  __shared__ float sA[TILE_M * FEAT];  // row-major tile [m*128 + k]   (8 KB)
  const int tid = threadIdx.x;

  // Stage W transposed into LDS (once per persistent block)
  for (int i = tid; i < FEAT * FEAT / 4; i += 256) {
    const float4 w = ((const float4*)W)[i];
    const int k = (i * 4) / FEAT;
    const int n = (i * 4) % FEAT;
    sW[(n + 0) * FEAT + k] = w.x;
    sW[(n + 1) * FEAT + k] = w.y;
    sW[(n + 2) * FEAT + k] = w.z;
    sW[(n + 3) * FEAT + k] = w.w;
  }

  const int wv = tid >> 5;      // wave id 0..7 -> N tile
  const int lane = tid & 31;
  const int mn = lane & 15;     // row (A) / col (B) index within tile
  const int half = lane >> 4;   // selects K pair within a K=4 slab
  const int nOff = wv * 16;
  const int nTiles = M / TILE_M;

  for (int t = blockIdx.x; t < nTiles; t += gridDim.x) {
    __syncthreads();  // previous iteration's compute done before sA overwrite
    const int m0 = t * TILE_M;
    for (int i = tid; i < TILE_M * FEAT / 4; i += 256)
      ((float4*)sA)[i] = ((const float4*)(A + (size_t)m0 * FEAT))[i];
    __syncthreads();  // sW + sA visible

    v8f acc = {};
#pragma unroll
    for (int k0 = 0; k0 < FEAT; k0 += 4) {
      const v2f av = *(const v2f*)&sA[mn * FEAT + k0 + 2 * half];
      const v2f bv = *(const v2f*)&sW[(nOff + mn) * FEAT + k0 + 2 * half];
      acc = wmma_f32(av, bv, acc);
    }

    // C/D layout: VGPR v -> row v + 8*half, col = lane&15 (+ tile offsets)
    float* Cp = C + (size_t)m0 * FEAT + nOff + mn;
    const float b = FINAL ? bias[nOff + mn] : 0.0f;
#pragma unroll
    for (int v = 0; v < 8; ++v) {
      float val = acc[v];
      if (ACC) val += Cp[(v + 8 * half) * FEAT];
      if (FINAL) {
        val += b;
        val = val > 0.0f ? val : 0.0f;
      }
      Cp[(v + 8 * half) * FEAT] = val;
    }
  }
}

// out[dst[e]] += vals[e] * H[src[e]]   -- one wave32 per edge, float4 per lane
__global__ void __launch_bounds__(256) spmm_scatter(
    const float* __restrict__ vals, const int* __restrict__ src,
    const int* __restrict__ dst, const float* __restrict__ H,
    float* __restrict__ out, int nEdges) {
  const int gid = blockIdx.x * 256 + threadIdx.x;
  const int e = gid >> 5;
  if (e >= nEdges) return;
  const int lane = gid & 31;
  const float v = vals[e];
  const size_t s = (size_t)src[e] * FEAT;
  const size_t d = (size_t)dst[e] * FEAT;
  const float4 h = *(const float4*)&H[s + lane * 4];
  float* o = &out[d + lane * 4];
  __hip_atomic_fetch_add(o + 0, v * h.x, __ATOMIC_RELAXED, __HIP_MEMORY_SCOPE_AGENT);
  __hip_atomic_fetch_add(o + 1, v * h.y, __ATOMIC_RELAXED, __HIP_MEMORY_SCOPE_AGENT);
  __hip_atomic_fetch_add(o + 2, v * h.z, __ATOMIC_RELAXED, __HIP_MEMORY_SCOPE_AGENT);
  __hip_atomic_fetch_add(o + 3, v * h.w, __ATOMIC_RELAXED, __HIP_MEMORY_SCOPE_AGENT);
}

// X = leaky(X + b) elementwise, float4 per thread
__global__ void __launch_bounds__(256) bias_leaky(float* __restrict__ X,
                                                  const float* __restrict__ b,
                                                  int total4) {
  const int i = blockIdx.x * 256 + threadIdx.x;
  if (i >= total4) return;
  float4 x = ((float4*)X)[i];
  const int col = (i * 4) & (FEAT - 1);
  const float4 bb = *(const float4*)&b[col];
  x.x += bb.x; x.y += bb.y; x.z += bb.z; x.w += bb.w;
  x.x = x.x >= 0.0f ? x.x : ALPHA * x.x;
  x.y = x.y >= 0.0f ? x.y : ALPHA * x.y;
  x.z = x.z >= 0.0f ? x.z : ALPHA * x.z;
  x.w = x.w >= 0.0f ? x.w : ALPHA * x.w;
  ((float4*)X)[i] = x;
}

extern "C" void kernel_launch(void* const* d_in, const int* in_sizes, int n_in,
                              void* d_out, int out_size, void* d_ws, size_t ws_size,
                              hipStream_t stream) {
  const float* ufea    = (const float*)d_in[0];
  const float* vfea    = (const float*)d_in[1];
  const int*   uv_rows = (const int*)d_in[2];
  const int*   uv_cols = (const int*)d_in[3];
  const float* uv_vals = (const float*)d_in[4];
  const float* vu_vals = (const float*)d_in[5];
  const float* W1 = (const float*)d_in[6];
  const float* b1 = (const float*)d_in[7];
  const float* W2 = (const float*)d_in[8];
  const float* b2 = (const float*)d_in[9];
  const float* W3 = (const float*)d_in[10];
  const float* b3 = (const float*)d_in[11];
  const float* W4 = (const float*)d_in[12];
  const float* b4 = (const float*)d_in[13];
  const float* Wu = (const float*)d_in[14];
  const float* bu = (const float*)d_in[15];
  const float* Wi = (const float*)d_in[16];
  const float* bi = (const float*)d_in[17];

  const int MU = in_sizes[0] / FEAT;  // N_USER
  const int MI = in_sizes[1] / FEAT;  // N_ITEM
  const int nE = in_sizes[4];
  const int MMAX = MU > MI ? MU : MI;

  float* B0 = (float*)d_ws;                       // MMAX x 128
  float* B1 = B0 + (size_t)MMAX * FEAT;           // MMAX x 128
  float* outU = (float*)d_out;
  float* outI = outU + (size_t)MU * FEAT;

  const dim3 blk(256);
  const int gGemm = 2500;  // persistent blocks, grid-stride over 16-row tiles
  const int gSpmm = (int)(((long long)nE * 32 + 255) / 256);
  const int gEwU = (MU * (FEAT / 4) + 255) / 256;
  const int gEwI = (MI * (FEAT / 4) + 255) / 256;
  const size_t rowsU = (size_t)MU * FEAT * sizeof(float);
  const size_t rowsI = (size_t)MI * FEAT * sizeof(float);

  // ---------------- user chain ----------------
  // t1 = ufea @ W1                          (users)
  gemm128<false, false><<<gGemm, blk, 0, stream>>>(ufea, W1, nullptr, B0, MU);
  // a1[item] = sum_e vu_vals[e] * t1[rows[e]]  (scatter by cols)
  hipMemsetAsync(B1, 0, rowsI, stream);
  spmm_scatter<<<gSpmm, blk, 0, stream>>>(vu_vals, uv_rows, uv_cols, B0, B1, nE);
  bias_leaky<<<gEwI, blk, 0, stream>>>(B1, b1, MI * (FEAT / 4));
  // t3 = a1 @ W3                            (items)
  gemm128<false, false><<<gGemm, blk, 0, stream>>>(B1, W3, nullptr, B0, MI);
  // a3[user] = sum_e uv_vals[e] * t3[cols[e]]  (scatter by rows)
  hipMemsetAsync(B1, 0, rowsU, stream);
  spmm_scatter<<<gSpmm, blk, 0, stream>>>(uv_vals, uv_cols, uv_rows, B0, B1, nE);
  bias_leaky<<<gEwU, blk, 0, stream>>>(B1, b3, MU * (FEAT / 4));
  // user = relu([a3, ufea] @ Wu + bu)  -> two K=128 passes
  gemm128<false, false><<<gGemm, blk, 0, stream>>>(B1, Wu, nullptr, outU, MU);
  gemm128<true, true><<<gGemm, blk, 0, stream>>>(ufea, Wu + FEAT * FEAT, bu, outU, MU);

  // ---------------- item chain ----------------
  // t2 = vfea @ W2                          (items)
  gemm128<false, false><<<gGemm, blk, 0, stream>>>(vfea, W2, nullptr, B0, MI);
  // a2[user] = sum_e uv_vals[e] * t2[cols[e]]
  hipMemsetAsync(B1, 0, rowsU, stream);
  spmm_scatter<<<gSpmm, blk, 0, stream>>>(uv_vals, uv_cols, uv_rows, B0, B1, nE);
  bias_leaky<<<gEwU, blk, 0, stream>>>(B1, b2, MU * (FEAT / 4));
  // t4 = a2 @ W4                            (users)
  gemm128<false, false><<<gGemm, blk, 0, stream>>>(B1, W4, nullptr, B0, MU);
  // a4[item] = sum_e vu_vals[e] * t4[rows[e]]
  hipMemsetAsync(B1, 0, rowsI, stream);
  spmm_scatter<<<gSpmm, blk, 0, stream>>>(vu_vals, uv_rows, uv_cols, B0, B1, nE);
  bias_leaky<<<gEwI, blk, 0, stream>>>(B1, b4, MI * (FEAT / 4));
  // item = relu([a4, vfea] @ Wi + bi)
  gemm128<false, false><<<gGemm, blk, 0, stream>>>(B1, Wi, nullptr, outI, MI);
  gemm128<true, true><<<gGemm, blk, 0, stream>>>(vfea, Wi + FEAT * FEAT, bi, outI, MI);
}